// DiffRenderer_90237262889751
// MI455X (gfx1250) — compile-verified
//
#include <hip/hip_runtime.h>
#include <math.h>

typedef __attribute__((ext_vector_type(2))) float v2f;
typedef __attribute__((ext_vector_type(8))) float v8f;

#define IMG 64
#define DST 64
#define DSH 61
#define H1  128
#define PIX_PER_BLK 8   // 8 wave32 waves per 256-thread block, 1 pixel per wave

// Fused differentiable-render kernel (transposed-WMMA orientation):
//  - fold z_shape into layer-1 bias (K: 64 -> 4)
//  - D = W1eff_tile(16x4) x feat(4x16) via v_wmma_f32_16x16x4_f32:
//      D rows = hidden units, D columns = points -> each lane owns ONE point,
//      so layer-2 needs only one shfl_xor(16) per 16-point subtile.
//  - per-pixel zero-crossing depth extraction in LDS
__global__ __launch_bounds__(256) void render_kernel(
    const float* __restrict__ z_shape, const float* __restrict__ z_extr,
    const float* __restrict__ W1, const float* __restrict__ b1,
    const float* __restrict__ W2, const float* __restrict__ b2,
    float* __restrict__ out)
{
    __shared__ float b1eff[H1];               // folded layer-1 bias (per batch)
    __shared__ float sdfL[PIX_PER_BLK][DST];  // per-wave SDF along the ray

    const int tid  = threadIdx.x;
    const int wv   = tid >> 5;
    const int lane = tid & 31;
    const int m16  = lane & 15;
    const int half = lane >> 4;   // 0: lanes 0-15, 1: lanes 16-31

    const int q    = blockIdx.x * PIX_PER_BLK + wv;  // global pixel id
    const int b    = q >> 12;                        // 4096 pixels per batch
    const int pix  = q & 4095;
    const int ypix = pix >> 6;
    const int xpix = pix & 63;

    // ---- fold z_shape into the layer-1 bias: b1eff[n] = b1[n] + sum_k z[k]*W1[3+k][n]
    // All 8 pixels in a block share the same batch (4096 % PIX_PER_BLK == 0).
    if (tid < H1) {
        float acc = b1[tid];
        const float* zrow = z_shape + b * DSH;
        for (int k = 0; k < DSH; ++k)
            acc = fmaf(zrow[k], W1[(3 + k) * H1 + tid], acc);
        b1eff[tid] = acc;
    }
    __syncthreads();

    // ---- per-batch extrinsics (uniform per wave)
    const float z0    = z_extr[b * 5 + 0];
    const float s_inv = z0;            // 1/s_obj
    const float s_obj = 1.0f / z0;
    const float tox   = z_extr[b * 5 + 1];
    const float toy   = z_extr[b * 5 + 2];
    const float toz   = z_extr[b * 5 + 3];
    const float ang   = 3.14159265358979323846f * z_extr[b * 5 + 4];
    const float cr = cosf(ang), sr = sinf(ang);
    const float bbd = toz + 2.5f;      // bbox mean depth: corner z-terms cancel exactly

    const float xc = ((float)xpix - 31.5f) * (1.0f / 70.0f);  // K_inv ray
    const float yc = ((float)ypix - 31.5f) * (1.0f / 70.0f);

    // ---- preload A tiles: W1eff is 4x128 (K x hidden); A_jt is 16(hidden) x 4(K).
    // Documented 16x4 A layout: lane holds M=lane%16; VGPR0=K0|K2 per half, VGPR1=K1|K3.
    v2f At[8];
#pragma unroll
    for (int jt = 0; jt < 8; ++jt) {
        const int n = jt * 16 + m16;                 // hidden unit (row M of tile jt)
        At[jt].x = half ? W1[2 * H1 + n] : W1[0 * H1 + n];
        At[jt].y = half ? b1eff[n]       : W1[1 * H1 + n];
    }
    // W2 rows owned by this lane in the D layout: M = jt*16 + half*8 + r, r=0..7
    float4 w2a[8], w2b[8];
#pragma unroll
    for (int jt = 0; jt < 8; ++jt) {
        const float* p = W2 + jt * 16 + half * 8;    // 32B aligned
        w2a[jt] = *(const float4*)(p);
        w2b[jt] = *(const float4*)(p + 4);
    }
    const float b2v = b2[0];

    // ---- 4 subtiles of 16 depth samples each cover this pixel's ray
#pragma unroll
    for (int st = 0; st < 4; ++st) {
        const int   d   = st * 16 + m16;             // this lane's point (column N)
        const float zsd = -1.0f + (float)d * (2.0f / 63.0f);
        const float Zv  = zsd * s_inv + bbd;
        const float u   = Zv * xc - tox;
        const float v   = Zv * yc - toy;
        const float w   = (Zv - 2.5f) - toz;
        const float Xx  = s_obj * (cr * u - sr * v);
        const float Xy  = s_obj * (sr * u + cr * v);
        const float Xz  = s_obj * w;

        // B (4x16 = K x points), assumed to mirror the A layout per lane half
        v2f Bv;
        Bv.x = half ? Xz   : Xx;
        Bv.y = half ? 1.0f : Xy;

        float acc = 0.0f;
#pragma unroll
        for (int jt = 0; jt < 8; ++jt) {
            v8f c = {};
            v8f h = __builtin_amdgcn_wmma_f32_16x16x4_f32(
                false, At[jt], false, Bv, (short)0, c, false, false);
            // layer 2 over the 8 hidden rows this lane holds for its point
            acc += fmaxf(h[0], 0.0f) * w2a[jt].x;
            acc += fmaxf(h[1], 0.0f) * w2a[jt].y;
            acc += fmaxf(h[2], 0.0f) * w2a[jt].z;
            acc += fmaxf(h[3], 0.0f) * w2a[jt].w;
            acc += fmaxf(h[4], 0.0f) * w2b[jt].x;
            acc += fmaxf(h[5], 0.0f) * w2b[jt].y;
            acc += fmaxf(h[6], 0.0f) * w2b[jt].z;
            acc += fmaxf(h[7], 0.0f) * w2b[jt].w;
        }
        // other 64 hidden units live in lane^16 (same point): single butterfly step
        const float tot = acc + __shfl_xor(acc, 16, 32);
        const float sdf = tanhf(tot + b2v);
        if (half == 0)
            sdfL[wv][st * 16 + m16] = sdf;
    }
    __syncthreads();

    // ---- zero-crossing depth extraction (2 depth samples per lane)
    float d1 = 100.0f, d2 = 100.0f, s1 = 0.0f, s2 = 0.0f;
#pragma unroll
    for (int h2 = 0; h2 < 2; ++h2) {
        const int   d    = lane + 32 * h2;
        const float sdfd = sdfL[wv][d];
        const float sdfn = (d < 63) ? sdfL[wv][d + 1] : 1.0f;   // forces zc1=false at d=63
        const float sdfp = (d > 0)  ? sdfL[wv][d - 1] : -1.0f;  // forces zc2=false at d=0
        const float zsd  = -1.0f + (float)d * (2.0f / 63.0f);
        const float Zd   = zsd * s_inv + bbd;
        const bool  zc1  = (sdfd > 0.0f) && (sdfn <= 0.0f) && (d < 63); // crossing starts at d
        const bool  zc2  = (sdfp > 0.0f) && (sdfd <= 0.0f) && (d > 0);  // crossing ends at d
        if (zc1) { d1 = fminf(d1, Zd); s1 += sdfd; }
        if (zc2) { d2 = fminf(d2, Zd); s2 += sdfd; }
    }
#pragma unroll
    for (int msk = 16; msk >= 1; msk >>= 1) {
        d1  = fminf(d1, __shfl_xor(d1, msk, 32));
        d2  = fminf(d2, __shfl_xor(d2, msk, 32));
        s1 += __shfl_xor(s1, msk, 32);
        s2 += __shfl_xor(s2, msk, 32);
    }
    if (lane == 0) {
        // any crossing => d1 holds a real Z (Z is bounded well below 100 for these inputs)
        const float occ   = (d1 < 99.0f) ? 1.0f : 0.0f;
        const float depth = occ * (d1 - s1 / (s2 - s1 - 1e-6f) * (d2 - d1));
        out[q] = depth;                    // depth_pred (BS,64,64,1)
        out[IMG * IMG * 4 + q] = occ;      // occ        (BS,64,64,1)
    }
}

extern "C" void kernel_launch(void* const* d_in, const int* in_sizes, int n_in,
                              void* d_out, int out_size, void* d_ws, size_t ws_size,
                              hipStream_t stream) {
    const float* z_shape = (const float*)d_in[0];
    const float* z_extr  = (const float*)d_in[1];
    const float* W1      = (const float*)d_in[2];
    const float* b1      = (const float*)d_in[3];
    const float* W2      = (const float*)d_in[4];
    const float* b2      = (const float*)d_in[5];
    float* out = (float*)d_out;
    (void)d_ws; (void)ws_size; (void)out_size; (void)n_in;

    const int BS = in_sizes[1] / 5;                         // 4
    const int nblocks = (BS * IMG * IMG) / PIX_PER_BLK;     // 2048 full blocks (EXEC all-1s)
    render_kernel<<<nblocks, 256, 0, stream>>>(z_shape, z_extr, W1, b1, W2, b2, out);
}